// Grouping_78408922956164
// MI455X (gfx1250) — compile-verified
//
#include <hip/hip_runtime.h>

typedef __attribute__((ext_vector_type(2))) float v2f;
typedef __attribute__((ext_vector_type(8))) float v8f;

namespace {
constexpr int kB = 8;     // batch
constexpr int kS = 8192;  // tokens per row
constexpr int kH = 512;   // hidden
constexpr int kG = 1024;  // max groups
constexpr int kWavesPerBlock = 8;
}

// One K=4 WMMA step: A = 16x4 one-hot segment indicator (exact in f32),
// B = 4x16 feats tile. Lane L supplies K = {0,1} (L<16) or {2,3} (L>=16)
// for both A and B per the CDNA5 16x16x4 f32 VGPR layout.
//
// Branchless: load indices are clamped to kS-1 (always-valid address), and
// the `tk < kS` test is folded into the indicator. Out-of-range-but-<S
// tokens self-mask via the sorted segment-id compare; the clamped stale
// value is finite, and 0.0 * finite == 0 in WMMA, so the contribution is
// exactly zero. No exec-mask manipulation in the hot loop.
__device__ __forceinline__ v8f seg_wmma_step(const int* __restrict__ segb,
                                             const float* __restrict__ featsb,
                                             int t, int khalf, int myGroup,
                                             int hcol, v8f acc) {
  const int tk0 = t + khalf;
  const int tk1 = tk0 + 1;
  const int c0 = (tk0 < kS) ? tk0 : (kS - 1);  // clamp address only
  const int c1 = (tk1 < kS) ? tk1 : (kS - 1);
  const int s0 = segb[c0];
  const int s1 = segb[c1];
  v2f a, b;
  a.x = ((s0 == myGroup) & (tk0 < kS)) ? 1.0f : 0.0f;
  a.y = ((s1 == myGroup) & (tk1 < kS)) ? 1.0f : 0.0f;
  b.x = featsb[c0 * kH + hcol];   // 32-bit index -> SGPR base + VGPR offset
  b.y = featsb[c1 * kH + hcol];
  return __builtin_amdgcn_wmma_f32_16x16x4_f32(
      /*neg_a=*/false, a, /*neg_b=*/false, b,
      /*c_mod=*/(short)0, acc, /*reuse_a=*/false, /*reuse_b=*/false);
}

__global__ __launch_bounds__(kWavesPerBlock * 32)
void seg_mean_wmma_kernel(const float* __restrict__ feats,
                          const int* __restrict__ seg,
                          float* __restrict__ grouped,
                          int* __restrict__ counts) {
  const int lane  = threadIdx.x & 31;
  const int wave  = threadIdx.x >> 5;
  const int htile = blockIdx.x * kWavesPerBlock + wave;
  const int g0    = blockIdx.y * 16;
  const int b     = blockIdx.z;
  const int h0    = htile * 16;

  const int*   segb   = seg   + (size_t)b * kS;
  const float* featsb = feats + (size_t)b * kS * kH;

  // lanes 0..16: lower_bound(segb, g0+lane); ids are sorted per batch row.
  const int target = g0 + (lane < 16 ? lane : 16);
  int lo = 0, hi = kS;
  while (lo < hi) {
    const int mid = (lo + hi) >> 1;
    if (segb[mid] < target) lo = mid + 1; else hi = mid;
  }
  const int start = __shfl(lo, 0);    // first token of group g0
  const int end   = __shfl(lo, 16);   // first token past group g0+15
  const int cnt   = __shfl_down(lo, 1) - lo;  // lanes 0..15: group lengths

  const int M       = lane & 15;            // row (A) / column (B,C,D)
  const int khalf   = (lane >> 4) << 1;     // K base: 0 or 2
  const int myGroup = g0 + M;
  const int hcol    = h0 + M;

  // Two accumulators to break the WMMA->WMMA RAW dependency chain.
  // Loop bound is wave-uniform (from shuffles), EXEC stays all-1s for WMMA.
  v8f acc0 = {};
  v8f acc1 = {};
  for (int t = start; t < end; t += 8) {
    int pf = t + khalf + 8;
    pf = (pf < kS - 1) ? pf : (kS - 1);
    __builtin_prefetch(&featsb[pf * kH + hcol], 0, 1);
    acc0 = seg_wmma_step(segb, featsb, t,     khalf, myGroup, hcol, acc0);
    acc1 = seg_wmma_step(segb, featsb, t + 4, khalf, myGroup, hcol, acc1);
  }

  // mean = sum / max(count, 1); row M of D lives in VGPR v = M - vbase.
  const float inv  = 1.0f / (float)(cnt < 1 ? 1 : cnt);  // lanes 0..15 valid
  const int  vbase = (lane >> 4) << 3;                   // 0 or 8
  float* outb = grouped + ((size_t)b * kG + g0) * kH + h0 + M;
#pragma unroll
  for (int v = 0; v < 8; ++v) {
    const int row   = vbase + v;
    const float s   = __shfl(inv, row);
    outb[(size_t)row * kH] = (acc0[v] + acc1[v]) * s;
  }

  if (htile == 0 && lane < 16) {
    counts[(size_t)b * kG + g0 + lane] = cnt;  // int32 bits into output tail
  }
}

extern "C" void kernel_launch(void* const* d_in, const int* in_sizes, int n_in,
                              void* d_out, int out_size, void* d_ws, size_t ws_size,
                              hipStream_t stream) {
  (void)in_sizes; (void)n_in; (void)out_size; (void)d_ws; (void)ws_size;
  const float* feats = (const float*)d_in[0];
  const int*   seg   = (const int*)d_in[1];
  float* grouped = (float*)d_out;
  int*   counts  = (int*)d_out + (size_t)kB * kG * kH;  // outputs concatenated

  dim3 grid(kH / 16 / kWavesPerBlock, kG / 16, kB);  // (4, 64, 8)
  seg_mean_wmma_kernel<<<grid, kWavesPerBlock * 32, 0, stream>>>(
      feats, seg, grouped, counts);
}